// JSDivergence_29815662968927
// MI455X (gfx1250) — compile-verified
//
#include <hip/hip_runtime.h>
#include <hip/hip_bf16.h>

typedef __attribute__((ext_vector_type(2))) float v2f;
typedef __attribute__((ext_vector_type(8))) float v8f;

#define D_DIM 256
#define NV 512   // rows of v_alpha (output columns j)
#define NT 512   // rows of t_alpha (output rows i)
#define EPS 1e-8f
#define PITCH 260  // LDS row pitch: 16B-aligned rows, 260%64=4 -> conflict-free column reads

// lnGamma(x) for x>0: shift by 4, Stirling at z=x+4>=4 (series through 1/(1260 z^5),
// abs err ~4e-8 at z=4 -> below f32 noise).  2x v_log_f32 + 1x v_rcp_f32.
__device__ __forceinline__ float lgamma_pos(float x) {
    float p  = x * (x + 1.f) * (x + 2.f) * (x + 3.f);
    float z  = x + 4.f;
    float r  = __builtin_amdgcn_rcpf(z);
    float r2 = r * r;
    float ser = r * (0.08333333333f + r2 * (-0.002777777778f + r2 * 0.0007936507937f));
    return (z - 0.5f) * __logf(z) - z + 0.91893853320467f + ser - __logf(p);
}

// digamma(x) for x>0: psi(x) = psi(x+4) - sum_{k=0..3} 1/(x+k), asymptotic at z>=4.
__device__ __forceinline__ float digamma_pos(float x) {
    float s = __builtin_amdgcn_rcpf(x) + __builtin_amdgcn_rcpf(x + 1.f)
            + __builtin_amdgcn_rcpf(x + 2.f) + __builtin_amdgcn_rcpf(x + 3.f);
    float z  = x + 4.f;
    float r  = __builtin_amdgcn_rcpf(z);
    float r2 = r * r;
    return __logf(z) - 0.5f * r
         - r2 * (0.08333333333f - r2 * (0.008333333333f - r2 * 0.003968253968f))
         - s;
}

// ---------------- Kernel 1: per-row stats + psi tables ----------------
// grid = NV+NT blocks, 32 threads (1 wave) per row.
// stat = { S, psi(S), 0.5*(lgamma(S) - sum lgamma) + 0.25*sum x*psi(x), 0 }
__global__ void jsd_rowstats(const float* __restrict__ V, const float* __restrict__ T,
                             float* __restrict__ PsiV, float* __restrict__ PsiT,
                             float4* __restrict__ statV, float4* __restrict__ statT) {
    int row  = blockIdx.x;
    int lane = threadIdx.x;
    const float* src; float* psid; float4* stat;
    if (row < NV) { src = V + row * D_DIM;        psid = PsiV + row * D_DIM;        stat = statV + row; }
    else { int r = row - NV; src = T + r * D_DIM; psid = PsiT + r * D_DIM;          stat = statT + r;   }

    float S = 0.f, G = 0.f, P = 0.f;
    #pragma unroll
    for (int k = 0; k < D_DIM / 32; ++k) {
        int d = lane + 32 * k;
        float x  = src[d] + EPS;
        float lg = lgamma_pos(x);
        float ps = digamma_pos(x);
        S += x; G += lg; P += x * ps;
        psid[d] = ps;
    }
    #pragma unroll
    for (int m = 16; m >= 1; m >>= 1) {
        S += __shfl_xor(S, m, 32);
        G += __shfl_xor(G, m, 32);
        P += __shfl_xor(P, m, 32);
    }
    if (lane == 0) {
        float4 o;
        o.x = S;
        o.y = digamma_pos(S);
        o.z = 0.5f * (lgamma_pos(S) - G) + 0.25f * P;
        o.w = 0.f;
        *stat = o;
    }
}

// ---------------- Kernel 2: fp32 WMMA cross-term GEMM + scalar epilogue ----------------
// One wave per 16x16 output tile.  C[i,j] = sum_d psi(t_id)*v_jd + t_id*psi(v_jd)
// via V_WMMA_F32_16X16X4_F32, 64 k-steps, 2 WMMAs each, accumulator chained.
// Writes base = 1 + lgamma(Sm) - r_v - r_t + 0.25*C + 0.25*(Sv-St)(psiSv-psiSt).
__global__ __launch_bounds__(32) void jsd_gemm_base(
        const float* __restrict__ V, const float* __restrict__ T,
        const float* __restrict__ PsiV, const float* __restrict__ PsiT,
        const float4* __restrict__ statV, const float4* __restrict__ statT,
        float* __restrict__ out) {
    int ti0  = blockIdx.y * 16;          // output rows  (t index i)
    int vj0  = blockIdx.x * 16;          // output cols  (v index j)
    int lane = threadIdx.x;
    int lm   = lane & 15;
    int half = lane >> 4;

    const float* psiT_row = PsiT + (ti0 + lm) * D_DIM;   // A1: 16x4 fragments, M = lm
    const float* t_row    = T    + (ti0 + lm) * D_DIM;   // A2
    const float* v_row    = V    + (vj0 + lm) * D_DIM;   // B1: B[k][n]=v[n][k] -> N = lm
    const float* psiV_row = PsiV + (vj0 + lm) * D_DIM;   // B2

    v8f acc = {0.f, 0.f, 0.f, 0.f, 0.f, 0.f, 0.f, 0.f};
    #pragma unroll 4
    for (int k0 = 0; k0 < D_DIM; k0 += 4) {
        // 32-bit A/B 16x4 layout: lanes 0-15 hold K=k0,k0+1; lanes 16-31 hold K=k0+2,k0+3
        int col = k0 + 2 * half;
        v2f a1 = *(const v2f*)(psiT_row + col);
        v2f b1 = *(const v2f*)(v_row    + col);  b1.x += EPS; b1.y += EPS;
        v2f a2 = *(const v2f*)(t_row    + col);  a2.x += EPS; a2.y += EPS;
        v2f b2 = *(const v2f*)(psiV_row + col);
        acc = __builtin_amdgcn_wmma_f32_16x16x4_f32(false, a1, false, b1, (short)0, acc, false, false);
        acc = __builtin_amdgcn_wmma_f32_16x16x4_f32(false, a2, false, b2, (short)0, acc, false, false);
    }

    // C/D layout: VGPR k, lanes 0-15 -> M=k, lanes 16-31 -> M=k+8; N = lane%16
    float4 sv = statV[vj0 + lm];
    #pragma unroll
    for (int k = 0; k < 8; ++k) {
        int i = ti0 + k + 8 * half;
        float4 st = statT[i];
        float Sm  = 0.5f * (sv.x + st.x);
        float base = 1.f + lgamma_pos(Sm) - sv.z - st.z
                   + 0.25f * acc[k]
                   + 0.25f * (sv.x - st.x) * (sv.y - st.y);
        out[i * NV + (vj0 + lm)] = base;
    }
}

// ---------------- Kernel 3: pairwise sum_d lgamma(0.5*(v+t)) ----------------
// 128 threads (4 waves) per 32x32 output tile; v/t rows staged in LDS (pitch 260).
// Each wave: 16x16 sub-tile, 8 independent lgamma accumulator chains per lane.
__global__ __launch_bounds__(128) void jsd_pairwise(
        const float* __restrict__ V, const float* __restrict__ T,
        float* __restrict__ out) {
    __shared__ float ve[32][PITCH];
    __shared__ float te[32][PITCH];
    int tid = threadIdx.x;
    int bj  = blockIdx.x * 32;   // v cols
    int bi  = blockIdx.y * 32;   // t rows

    // stage 32 rows x 256 floats of each input (eps pre-added) with float4 loads
    for (int idx = tid; idx < 32 * (D_DIM / 4); idx += 128) {
        int row = idx >> 6;
        int c4  = (idx & 63) * 4;
        float4 v4 = *(const float4*)(V + (bj + row) * D_DIM + c4);
        float4 t4 = *(const float4*)(T + (bi + row) * D_DIM + c4);
        v4.x += EPS; v4.y += EPS; v4.z += EPS; v4.w += EPS;
        t4.x += EPS; t4.y += EPS; t4.z += EPS; t4.w += EPS;
        *(float4*)(&ve[row][c4]) = v4;
        *(float4*)(&te[row][c4]) = t4;
    }
    __syncthreads();

    int lane = tid & 31;
    int wid  = tid >> 5;
    int wx   = wid & 1, wy = wid >> 1;     // wave's 16x16 quadrant in the 32x32 tile
    int lm   = lane & 15, half = lane >> 4;

    const float* vcol = &ve[wx * 16 + lm][0];            // distinct banks across lanes
    const float* trow = &te[wy * 16 + 8 * half][0];      // broadcast reads

    float accs[8] = {0.f, 0.f, 0.f, 0.f, 0.f, 0.f, 0.f, 0.f};
    for (int d = 0; d < D_DIM; ++d) {
        float tv = vcol[d];
        #pragma unroll
        for (int k = 0; k < 8; ++k) {
            float m = 0.5f * (tv + trow[k * PITCH + d]);
            accs[k] += lgamma_pos(m);
        }
    }

    int j = bj + wx * 16 + lm;
    #pragma unroll
    for (int k = 0; k < 8; ++k) {
        int i = bi + wy * 16 + 8 * half + k;
        out[i * NV + j] -= accs[k];        // base (kernel 2) minus pairwise lgamma sum
    }
}

extern "C" void kernel_launch(void* const* d_in, const int* in_sizes, int n_in,
                              void* d_out, int out_size, void* d_ws, size_t ws_size,
                              hipStream_t stream) {
    const float* V = (const float*)d_in[0];   // v_alpha [512,256]
    const float* T = (const float*)d_in[1];   // t_alpha [512,256]
    float* out = (float*)d_out;               // [512,512] f32

    float*  PsiV  = (float*)d_ws;                          // 512*256 f32
    float*  PsiT  = PsiV + NV * D_DIM;                     // 512*256 f32
    float4* statV = (float4*)(PsiT + NT * D_DIM);          // 512 float4 (16B aligned @1MB)
    float4* statT = statV + NV;                            // 512 float4

    jsd_rowstats<<<dim3(NV + NT), dim3(32), 0, stream>>>(V, T, PsiV, PsiT, statV, statT);
    jsd_gemm_base<<<dim3(NV / 16, NT / 16), dim3(32), 0, stream>>>(V, T, PsiV, PsiT, statV, statT, out);
    jsd_pairwise<<<dim3(NV / 32, NT / 32), dim3(128), 0, stream>>>(V, T, out);
}